// ParallelMamba_65489661330035
// MI455X (gfx1250) — compile-verified
//
#include <hip/hip_runtime.h>
#include <math.h>

typedef __attribute__((ext_vector_type(2))) float v2f;
typedef __attribute__((ext_vector_type(8))) float v8f;

#define NM 4        // n_mambas
#define DM 1024     // d_model
#define DI 2048     // d_inner
#define DS 16       // d_state
#define DTR 64      // dt_rank
#define BB 2        // batch
#define LC 1024     // seqlen per mamba (4096/4)
#define RR (BB*LC)  // 2048 GEMM rows per mamba
#define XD 96       // dt_rank + 2*d_state

// workspace layout (float offsets)
#define OFF_XZ   ((size_t)0)
#define SZ_XZ    ((size_t)NM*RR*2*DI)       // 33,554,432
#define OFF_CONV (OFF_XZ + SZ_XZ)
#define SZ_CONV  ((size_t)NM*RR*DI)         // 16,777,216
#define OFF_XDBL (OFF_CONV + SZ_CONV)
#define SZ_XDBL  ((size_t)NM*RR*XD)         //    786,432
#define OFF_DT   (OFF_XDBL + SZ_XDBL)
#define SZ_DT    ((size_t)NM*RR*DI)         // 16,777,216 (dt in, y out, in-place)
#define WS_FLOATS (OFF_DT + SZ_DT)

__device__ __forceinline__ v8f wmma_f32(v2f a, v2f b, v8f c) {
  // D = A(16x4 f32) * B(4x16 f32) + C ; emits v_wmma_f32_16x16x4_f32
  return __builtin_amdgcn_wmma_f32_16x16x4_f32(false, a, false, b, (short)0, c,
                                               false, false);
}

// ---------------- K1: in_proj  xz[n][r][0..4096) = x_gather[r] @ Wi[n]^T ---
__global__ __launch_bounds__(32)
void k_in_proj(const float* __restrict__ x, const float* __restrict__ Wi,
               float* __restrict__ xz)
{
  const int lane = threadIdx.x;
  const int n    = blockIdx.z;
  const int mt   = blockIdx.y;          // 128 tiles of 16 rows
  const int ncol = blockIdx.x * 64;     // 64 strips of 64 cols
  const float* W = Wi + (size_t)n * (2*DI) * DM;

  // row r -> source x row b*4096 + 4*t + n (interleave gather)
  int r    = mt * 16 + (lane & 15);
  int xrow = ((r >> 10) << 12) + ((r & 1023) << 2) + n;
  const float* arow = x + (size_t)xrow * DM + ((lane >> 4) << 1);

  v8f z = {};
  v8f acc[4]; acc[0] = z; acc[1] = z; acc[2] = z; acc[3] = z;
  const int koff = (lane >> 4) << 1;
  for (int k0 = 0; k0 < DM; k0 += 4) {
    v2f a; a.x = arow[k0]; a.y = arow[k0 + 1];
#pragma unroll
    for (int j = 0; j < 4; ++j) {
      const float* bp = W + (size_t)(ncol + 16*j + (lane & 15)) * DM + k0 + koff;
      v2f b; b.x = bp[0]; b.y = bp[1];
      acc[j] = wmma_f32(a, b, acc[j]);
    }
  }
  float* out = xz + (size_t)n * RR * (2*DI);
  int mbase = mt * 16 + ((lane >> 4) << 3);
  int nn    = ncol + (lane & 15);
#pragma unroll
  for (int jt = 0; jt < 4; ++jt)
#pragma unroll
    for (int j = 0; j < 8; ++j)
      out[(size_t)(mbase + j) * (2*DI) + nn + 16*jt] = acc[jt][j];
}

// ---------------- K2: depthwise causal conv1d + SiLU ----------------------
__global__ __launch_bounds__(256)
void k_conv_silu(const float* __restrict__ xz, const float* __restrict__ cw,
                 const float* __restrict__ cb, float* __restrict__ conv)
{
  size_t idx = (size_t)blockIdx.x * 256 + threadIdx.x;   // NM*BB*LC*DI = 1<<24
  int d = idx & (DI - 1);
  int t = (int)((idx >> 11) & (LC - 1));
  int b = (int)((idx >> 21) & 1);
  int n = (int)(idx >> 22);
  const float* xc = xz + (((size_t)n * BB + b) * LC) * (size_t)(2*DI) + d;
  const float* w  = cw + ((size_t)n * DI + d) * 4;
  float acc = cb[n * DI + d];
  if (t >= 3) acc += w[0] * xc[(size_t)(t - 3) * (2*DI)];
  if (t >= 2) acc += w[1] * xc[(size_t)(t - 2) * (2*DI)];
  if (t >= 1) acc += w[2] * xc[(size_t)(t - 1) * (2*DI)];
  acc += w[3] * xc[(size_t)t * (2*DI)];
  conv[idx] = acc / (1.f + __expf(-acc));                // silu
}

// ---------------- K3: x_dbl = conv @ Wx^T  (N=96) -------------------------
__global__ __launch_bounds__(32)
void k_xdbl(const float* __restrict__ conv, const float* __restrict__ Wx,
            float* __restrict__ xdbl)
{
  const int lane = threadIdx.x;
  const int n  = blockIdx.z;
  const int mt = blockIdx.y;
  const int ncol = blockIdx.x * 16;     // 6 tiles
  const float* A = conv + (size_t)n * RR * DI;
  const float* W = Wx + (size_t)n * XD * DI;
  const int koff = (lane >> 4) << 1;
  const float* arow = A + (size_t)(mt * 16 + (lane & 15)) * DI + koff;
  const float* brow = W + (size_t)(ncol + (lane & 15)) * DI + koff;
  v8f acc = {};
  for (int k0 = 0; k0 < DI; k0 += 4) {
    v2f a; a.x = arow[k0]; a.y = arow[k0 + 1];
    v2f b; b.x = brow[k0]; b.y = brow[k0 + 1];
    acc = wmma_f32(a, b, acc);
  }
  float* out = xdbl + (size_t)n * RR * XD;
  int mbase = mt * 16 + ((lane >> 4) << 3);
  int nn = ncol + (lane & 15);
#pragma unroll
  for (int j = 0; j < 8; ++j)
    out[(size_t)(mbase + j) * XD + nn] = acc[j];
}

// ---------------- K4: dt = softplus(x_dbl[:, :64] @ Wd^T + bias) ----------
__global__ __launch_bounds__(32)
void k_dt(const float* __restrict__ xdbl, const float* __restrict__ Wd,
          const float* __restrict__ db, float* __restrict__ dt)
{
  const int lane = threadIdx.x;
  const int n  = blockIdx.z;
  const int mt = blockIdx.y;
  const int ncol = blockIdx.x * 64;     // 32 strips over N=2048
  const float* A = xdbl + (size_t)n * RR * XD;
  const float* W = Wd + (size_t)n * DI * DTR;
  const int koff = (lane >> 4) << 1;
  const float* arow = A + (size_t)(mt * 16 + (lane & 15)) * XD + koff;
  v8f z = {};
  v8f acc[4]; acc[0] = z; acc[1] = z; acc[2] = z; acc[3] = z;
  for (int k0 = 0; k0 < DTR; k0 += 4) {
    v2f a; a.x = arow[k0]; a.y = arow[k0 + 1];
#pragma unroll
    for (int j = 0; j < 4; ++j) {
      const float* bp = W + (size_t)(ncol + 16*j + (lane & 15)) * DTR + k0 + koff;
      v2f b; b.x = bp[0]; b.y = bp[1];
      acc[j] = wmma_f32(a, b, acc[j]);
    }
  }
  float* out = dt + (size_t)n * RR * DI;
  const float* bias = db + (size_t)n * DI;
  int mbase = mt * 16 + ((lane >> 4) << 3);
  int nn = ncol + (lane & 15);
#pragma unroll
  for (int jt = 0; jt < 4; ++jt)
#pragma unroll
    for (int j = 0; j < 8; ++j) {
      float v = acc[jt][j] + bias[nn + 16*jt];
      float sp = (v > 20.f) ? v : __logf(1.f + __expf(v));  // softplus
      out[(size_t)(mbase + j) * DI + nn + 16*jt] = sp;
    }
}

// ---------------- K5: selective scan + D-skip + SiLU gate (y over dt) -----
#define TCH 64
__global__ __launch_bounds__(256)
void k_scan(const float* __restrict__ xz, const float* __restrict__ conv,
            const float* __restrict__ xdbl, const float* __restrict__ A_log,
            const float* __restrict__ Dp, float* dty)
{
  __shared__ float sB[TCH][DS];
  __shared__ float sC[TCH][DS];
  const int tid = threadIdx.x;
  const int nb = blockIdx.y, n = nb >> 1, b = nb & 1;
  const int d = blockIdx.x * 256 + tid;

  float* dtp = dty + (size_t)n * RR * DI + (size_t)b * LC * DI + d;  // dt in / y out
  const float* cvp = conv + (size_t)n * RR * DI + (size_t)b * LC * DI + d;
  const float* zp  = xz + (size_t)n * RR * (2*DI) + (size_t)b * LC * (2*DI) + DI + d;
  const float* xdp = xdbl + (size_t)n * RR * XD + (size_t)b * LC * XD;
  const float* alp = A_log + ((size_t)n * DI + d) * DS;

  float Arow[DS], h[DS];
#pragma unroll
  for (int s = 0; s < DS; ++s) { Arow[s] = -__expf(alp[s]); h[s] = 0.f; }
  const float Dd = Dp[n * DI + d];

  for (int tc = 0; tc < LC; tc += TCH) {
    __syncthreads();
    for (int i = tid; i < TCH * 2 * DS; i += 256) {      // stage B_t, C_t
      int tt = i >> 5, j = i & 31;
      float v = xdp[(size_t)(tc + tt) * XD + DTR + j];
      if (j < DS) sB[tt][j] = v; else sC[tt][j - DS] = v;
    }
    __syncthreads();
    for (int tt = 0; tt < TCH; ++tt) {
      int t = tc + tt;
      float dtv = dtp[(size_t)t * DI];
      float xv  = cvp[(size_t)t * DI];
      float zv  = zp[(size_t)t * (2*DI)];
      float du  = dtv * xv;
      float y = 0.f;
#pragma unroll
      for (int s = 0; s < DS; ++s) {
        float dA = __expf(dtv * Arow[s]);
        h[s] = dA * h[s] + du * sB[tt][s];
        y += h[s] * sC[tt][s];
      }
      float yf = (y + xv * Dd) * (zv / (1.f + __expf(-zv)));
      dtp[(size_t)t * DI] = yf;                           // overwrite dead dt
    }
  }
}

// ---------------- K6: out = y @ Wo^T, de-interleaved store ----------------
__global__ __launch_bounds__(32)
void k_out_proj(const float* __restrict__ y, const float* __restrict__ Wo,
                float* __restrict__ out)
{
  const int lane = threadIdx.x;
  const int n  = blockIdx.z;
  const int mt = blockIdx.y;
  const int ncol = blockIdx.x * 64;     // 16 strips over N=1024
  const float* A = y + (size_t)n * RR * DI;
  const float* W = Wo + (size_t)n * DM * DI;
  const int koff = (lane >> 4) << 1;
  const float* arow = A + (size_t)(mt * 16 + (lane & 15)) * DI + koff;
  v8f z = {};
  v8f acc[4]; acc[0] = z; acc[1] = z; acc[2] = z; acc[3] = z;
  for (int k0 = 0; k0 < DI; k0 += 4) {
    v2f a; a.x = arow[k0]; a.y = arow[k0 + 1];
#pragma unroll
    for (int j = 0; j < 4; ++j) {
      const float* bp = W + (size_t)(ncol + 16*j + (lane & 15)) * DI + k0 + koff;
      v2f b; b.x = bp[0]; b.y = bp[1];
      acc[j] = wmma_f32(a, b, acc[j]);
    }
  }
  // out[b][t][n*DM + m] : row r = b*LC+t -> flat r*4096 + n*1024 + col
  int mbase = mt * 16 + ((lane >> 4) << 3);
  int nn = ncol + (lane & 15);
#pragma unroll
  for (int jt = 0; jt < 4; ++jt)
#pragma unroll
    for (int j = 0; j < 8; ++j)
      out[(size_t)(mbase + j) * (NM*DM) + n*DM + nn + 16*jt] = acc[jt][j];
}

extern "C" void kernel_launch(void* const* d_in, const int* in_sizes, int n_in,
                              void* d_out, int out_size, void* d_ws, size_t ws_size,
                              hipStream_t stream)
{
  (void)in_sizes; (void)n_in; (void)out_size;
  if (ws_size < WS_FLOATS * sizeof(float)) return;   // need ~272 MB scratch

  const float* x  = (const float*)d_in[0];
  const float* wi = (const float*)d_in[1];
  const float* cw = (const float*)d_in[2];
  const float* cb = (const float*)d_in[3];
  const float* wx = (const float*)d_in[4];
  const float* wd = (const float*)d_in[5];
  const float* db = (const float*)d_in[6];
  const float* al = (const float*)d_in[7];
  const float* Dp = (const float*)d_in[8];
  const float* wo = (const float*)d_in[9];
  float* out = (float*)d_out;

  float* ws   = (float*)d_ws;
  float* xz   = ws + OFF_XZ;
  float* conv = ws + OFF_CONV;
  float* xdbl = ws + OFF_XDBL;
  float* dty  = ws + OFF_DT;

  k_in_proj  <<<dim3(64, 128, NM), 32, 0, stream>>>(x, wi, xz);
  k_conv_silu<<<dim3((NM*BB*LC*DI)/256), 256, 0, stream>>>(xz, cw, cb, conv);
  k_xdbl     <<<dim3(6, 128, NM), 32, 0, stream>>>(conv, wx, xdbl);
  k_dt       <<<dim3(32, 128, NM), 32, 0, stream>>>(xdbl, wd, db, dty);
  k_scan     <<<dim3(DI/256, NM*BB), 256, 0, stream>>>(xz, conv, xdbl, al, Dp, dty);
  k_out_proj <<<dim3(16, 128, NM), 32, 0, stream>>>(dty, wo, out);
}